// GraphSAGEPredictor_4183298146581
// MI455X (gfx1250) — compile-verified
//
#include <hip/hip_runtime.h>

#define N_NODES   100000
#define N_EDGES   1200000
#define N_GRAPHS  1024
#define IN_FEATS  64
#define HID       64
#define BN_EPS    1e-5f

typedef __attribute__((ext_vector_type(2))) float v2f;
typedef __attribute__((ext_vector_type(8))) float v8f;

// ---------------------------------------------------------------------------
// Zero workspace (agg | deg | hg are laid out contiguously)
// ---------------------------------------------------------------------------
__global__ void zero_kernel(float* __restrict__ p, int n) {
    int i = blockIdx.x * blockDim.x + threadIdx.x;
    int stride = gridDim.x * blockDim.x;
    for (; i < n; i += stride) p[i] = 0.0f;
}

// ---------------------------------------------------------------------------
// Edge scatter-add: one wave32 per edge, float2 per lane (coalesced 256B row).
// Lane 0 also accumulates the degree. All data is L2-resident (feats = 25.6MB).
// ---------------------------------------------------------------------------
__global__ void __launch_bounds__(256)
edge_agg_kernel(const float* __restrict__ feats,
                const int*   __restrict__ src,
                const int*   __restrict__ dst,
                float*       __restrict__ agg,
                float*       __restrict__ deg) {
    int tid  = blockIdx.x * blockDim.x + threadIdx.x;
    int e    = tid >> 5;
    int lane = tid & 31;
    if (e >= N_EDGES) return;
    int s = src[e];
    int d = dst[e];
    float2 f = reinterpret_cast<const float2*>(feats)[(size_t)s * 32 + lane];
    float* a = agg + (size_t)d * 64 + lane * 2;
    atomicAdd(a,     f.x);
    atomicAdd(a + 1, f.y);
    if (lane == 0) atomicAdd(deg + d, 1.0f);
}

// ---------------------------------------------------------------------------
// Fused SAGE layer + per-graph max readout.
// One wave per 16x16 output tile of h = relu(feats@Wself + (agg/deg)@Wneigh + b).
// Uses V_WMMA_F32_16X16X4_F32 (exact fp32), K=64 in steps of 4 for each GEMM,
// both accumulated into one v8f. h is never stored: result goes straight into
// hg via uint atomicMax (valid: post-ReLU values are >= 0).
// ---------------------------------------------------------------------------
__global__ void __launch_bounds__(256)
sage_wmma_kernel(const float* __restrict__ feats,
                 const float* __restrict__ agg,
                 const float* __restrict__ deg,
                 const float* __restrict__ Wself,
                 const float* __restrict__ Wneigh,
                 const float* __restrict__ bneigh,
                 const int*   __restrict__ graph_ids,
                 unsigned int* __restrict__ hg_bits) {
    const int wave = (blockIdx.x * blockDim.x + threadIdx.x) >> 5;
    const int lane = threadIdx.x & 31;
    const int NT   = HID / 16;               // 4 N-tiles
    const int mt   = wave / NT;
    const int nt   = wave % NT;
    if (mt * 16 >= N_NODES) return;          // wave-uniform guard (EXEC stays full)

    const int m0   = mt * 16;
    const int n0   = nt * 16;
    const int lr   = lane & 15;              // M (for A) / N (for B,C,D)
    const int koff = (lane >> 4) << 1;       // 0 for lanes 0-15, 2 for lanes 16-31
    const int rowA = m0 + lr;
    const int col  = n0 + lr;

    const float invd = 1.0f / fmaxf(deg[rowA], 1.0f);

    const float2* fA = reinterpret_cast<const float2*>(feats) + (size_t)rowA * 32;
    const float2* gA = reinterpret_cast<const float2*>(agg)   + (size_t)rowA * 32;

    v8f c = {0.f, 0.f, 0.f, 0.f, 0.f, 0.f, 0.f, 0.f};

    // self GEMM: feats @ Wself
    #pragma unroll
    for (int k = 0; k < IN_FEATS; k += 4) {
        float2 t = fA[(k + koff) >> 1];
        v2f a; a.x = t.x; a.y = t.y;
        v2f b; b.x = Wself[(k + koff)     * HID + col];
               b.y = Wself[(k + koff + 1) * HID + col];
        c = __builtin_amdgcn_wmma_f32_16x16x4_f32(false, a, false, b,
                                                  (short)0, c, false, false);
    }
    // neighbor GEMM: (agg/deg) @ Wneigh  (1/deg folded into A load)
    #pragma unroll
    for (int k = 0; k < IN_FEATS; k += 4) {
        float2 t = gA[(k + koff) >> 1];
        v2f a; a.x = t.x * invd; a.y = t.y * invd;
        v2f b; b.x = Wneigh[(k + koff)     * HID + col];
               b.y = Wneigh[(k + koff + 1) * HID + col];
        c = __builtin_amdgcn_wmma_f32_16x16x4_f32(false, a, false, b,
                                                  (short)0, c, false, false);
    }

    const float bias  = bneigh[col];
    const int   rbase = (lane < 16) ? 0 : 8;  // C/D layout: VGPR j -> row j or j+8
    #pragma unroll
    for (int j = 0; j < 8; ++j) {
        int   r   = m0 + rbase + j;
        float v   = fmaxf(c[j] + bias, 0.0f);
        int   gid = graph_ids[r];
        atomicMax(&hg_bits[(size_t)gid * HID + col], __float_as_uint(v));
    }
}

// ---------------------------------------------------------------------------
// MLP head: one 128-thread block per graph; 64->128 (BN,ReLU) -> 64 (BN,ReLU)
// -> 1, staged through LDS. 17 MFLOP total -- negligible.
// ---------------------------------------------------------------------------
__global__ void __launch_bounds__(128)
mlp_kernel(const float* __restrict__ hg,
           const float* __restrict__ W1,  const float* __restrict__ b1,
           const float* __restrict__ g1,  const float* __restrict__ be1,
           const float* __restrict__ rm1, const float* __restrict__ rv1,
           const float* __restrict__ W2,  const float* __restrict__ b2,
           const float* __restrict__ g2,  const float* __restrict__ be2,
           const float* __restrict__ rm2, const float* __restrict__ rv2,
           const float* __restrict__ W3,  const float* __restrict__ b3,
           float* __restrict__ out) {
    __shared__ float s_in[64];
    __shared__ float s_x1[128];
    __shared__ float s_x2[64];
    const int g = blockIdx.x;
    const int t = threadIdx.x;

    if (t < 64) s_in[t] = hg[(size_t)g * 64 + t];
    __syncthreads();

    // layer 1: 64 -> 128, ReLU then eval-BN
    float acc = b1[t];
    #pragma unroll 8
    for (int k = 0; k < 64; ++k) acc += s_in[k] * W1[k * 128 + t];
    acc = fmaxf(acc, 0.0f);
    acc = (acc - rm1[t]) * rsqrtf(rv1[t] + BN_EPS) * g1[t] + be1[t];
    s_x1[t] = acc;
    __syncthreads();

    // layer 2: 128 -> 64, ReLU then eval-BN; premultiply by W3 for reduction
    if (t < 64) {
        float a2 = b2[t];
        #pragma unroll 8
        for (int k = 0; k < 128; ++k) a2 += s_x1[k] * W2[k * 64 + t];
        a2 = fmaxf(a2, 0.0f);
        a2 = (a2 - rm2[t]) * rsqrtf(rv2[t] + BN_EPS) * g2[t] + be2[t];
        s_x2[t] = a2 * W3[t];
    }
    __syncthreads();

    if (t == 0) {
        float s = b3[0];
        #pragma unroll
        for (int k = 0; k < 64; ++k) s += s_x2[k];
        out[g] = s;
    }
}

// ---------------------------------------------------------------------------
extern "C" void kernel_launch(void* const* d_in, const int* in_sizes, int n_in,
                              void* d_out, int out_size, void* d_ws, size_t ws_size,
                              hipStream_t stream) {
    const float* feats     = (const float*)d_in[0];
    const int*   src       = (const int*)  d_in[1];
    const int*   dst       = (const int*)  d_in[2];
    const int*   graph_ids = (const int*)  d_in[3];
    const float* Wself     = (const float*)d_in[4];
    const float* Wneigh    = (const float*)d_in[5];
    const float* bneigh    = (const float*)d_in[6];
    const float* W1        = (const float*)d_in[7];
    const float* b1        = (const float*)d_in[8];
    const float* g1        = (const float*)d_in[9];
    const float* be1       = (const float*)d_in[10];
    const float* rm1       = (const float*)d_in[11];
    const float* rv1       = (const float*)d_in[12];
    const float* W2        = (const float*)d_in[13];
    const float* b2        = (const float*)d_in[14];
    const float* g2        = (const float*)d_in[15];
    const float* be2       = (const float*)d_in[16];
    const float* rm2       = (const float*)d_in[17];
    const float* rv2       = (const float*)d_in[18];
    const float* W3        = (const float*)d_in[19];
    const float* b3        = (const float*)d_in[20];

    float* ws  = (float*)d_ws;
    float* agg = ws;                                   // N_NODES*64 floats
    float* deg = agg + (size_t)N_NODES * 64;           // N_NODES floats
    float* hg  = deg + N_NODES;                        // N_GRAPHS*64 floats
    const int zero_n = N_NODES * 64 + N_NODES + N_GRAPHS * 64;

    zero_kernel<<<4096, 256, 0, stream>>>(ws, zero_n);

    // one wave32 per edge: 1.2M * 32 threads
    const int edge_blocks = (N_EDGES * 32) / 256;      // 150000
    edge_agg_kernel<<<edge_blocks, 256, 0, stream>>>(feats, src, dst, agg, deg);

    // 6250 M-tiles * 4 N-tiles = 25000 waves = 3125 blocks of 8 waves
    const int tiles = (N_NODES / 16) * (HID / 16);
    sage_wmma_kernel<<<(tiles * 32) / 256, 256, 0, stream>>>(
        feats, agg, deg, Wself, Wneigh, bneigh, graph_ids, (unsigned int*)hg);

    mlp_kernel<<<N_GRAPHS, 128, 0, stream>>>(hg, W1, b1, g1, be1, rm1, rv1,
                                             W2, b2, g2, be2, rm2, rv2,
                                             W3, b3, (float*)d_out);
}